// FixedAverageModel_30288109372028
// MI455X (gfx1250) — compile-verified
//
#include <hip/hip_runtime.h>

#define S_LEN 512
#define D_DIM 768
#define NLAB  9
#define MW    256
#define WBLK  16
#define BLOCK 256

typedef __attribute__((ext_vector_type(2))) float v2f;
typedef __attribute__((ext_vector_type(8))) float v8f;

// lower_bound on sorted int array (uniform across the block -> scalar path)
__device__ __forceinline__ int lbound(const int* __restrict__ a, int n, int v) {
    int lo = 0, hi = n;
    while (lo < hi) {
        int m = (lo + hi) >> 1;
        if (a[m] < v) lo = m + 1; else hi = m;
    }
    return lo;
}

__global__ __launch_bounds__(BLOCK)
void fused_align_blend_gemm(const float* __restrict__ hh,   // (B,S,D) hing hidden
                            const float* __restrict__ hr,   // (B,S,D) rob hidden
                            const int*   __restrict__ wih,  // (B,S) hing word ids (sorted)
                            const int*   __restrict__ wir,  // (B,S) rob word ids (sorted)
                            const float* __restrict__ W,    // (D, NLAB)
                            const float* __restrict__ bias, // (NLAB)
                            float* __restrict__ logits,     // (B, MW, NLAB)
                            float* __restrict__ alpha)      // (B, MW)
{
    __shared__ __align__(16) float s_bl[WBLK * D_DIM];   // blended tile, 48 KB
    __shared__ __align__(16) float s_tmp[WBLK * D_DIM];  // accumulator, then W^T padded, 48 KB
    __shared__ float s_cnt[WBLK];

    const int tid = threadIdx.x;
    const int bb  = blockIdx.y;          // batch
    const int w0  = blockIdx.x * WBLK;   // first word of this tile

    const int* idh = wih + (size_t)bb * S_LEN;
    const int* idr = wir + (size_t)bb * S_LEN;

    // ---- two segmented-mean passes: 0 = hing, 1 = rob -----------------
    for (int pass = 0; pass < 2; ++pass) {
        const int*   ids = pass ? idr : idh;
        const float* h   = (pass ? hr : hh) + (size_t)bb * S_LEN * D_DIM;

        // zero the accumulator tile
        float4* t4 = (float4*)s_tmp;
        for (int i = tid; i < (WBLK * D_DIM) / 4; i += BLOCK)
            t4[i] = make_float4(0.f, 0.f, 0.f, 0.f);

        // per-word counts via adjacent binary searches (ids sorted)
        if (tid < WBLK) {
            int c0 = lbound(ids, S_LEN, w0 + tid);
            int c1 = lbound(ids, S_LEN, w0 + tid + 1);
            s_cnt[tid] = (float)(c1 - c0);
        }
        __syncthreads();

        // contiguous token range feeding this word tile
        const int lo = lbound(ids, S_LEN, w0);
        const int hi = lbound(ids, S_LEN, w0 + WBLK);

        // 192 lanes each own one float4 column; tokens processed in order,
        // so each LDS address is touched by exactly one thread -> no atomics
        if (tid < D_DIM / 4) {
            for (int s = lo; s < hi; ++s) {
                int row = ids[s] - w0;
                const float4* src = (const float4*)(h + (size_t)s * D_DIM);
                float4 v = src[tid];
                float4* dst = (float4*)(s_tmp + row * D_DIM) + tid;
                float4 d = *dst;
                d.x += v.x; d.y += v.y; d.z += v.z; d.w += v.w;
                *dst = d;
            }
        }
        __syncthreads();

        // mean * 0.5, blend into s_bl
        for (int i = tid; i < WBLK * D_DIM; i += BLOCK) {
            int   w = i / D_DIM;
            float c = s_cnt[w];
            float m = (c > 0.f) ? (0.5f * s_tmp[i] / c) : 0.f;
            if (pass == 0) s_bl[i] = m;
            else           s_bl[i] += m;
        }
        __syncthreads();
    }

    // ---- stage W transposed + zero-padded (9 -> 16 cols) into s_tmp ---
    // s_tmp[n*D + k] = (n < 9) ? W[k*9 + n] : 0
    for (int i = tid; i < WBLK * D_DIM; i += BLOCK) {
        int n = i / D_DIM;
        int k = i - n * D_DIM;
        s_tmp[i] = (n < NLAB) ? W[(size_t)k * NLAB + n] : 0.f;
    }
    __syncthreads();

    // ---- 16x16 fp32 WMMA GEMM over K = 768 (wave 0 only) --------------
    if (tid < 32) {
        const int lane = tid;
        const int lm   = lane & 15;          // A: row M / B: col N
        const int kof  = (lane >> 4) << 1;   // lanes 16-31 carry K+2,K+3

        const float* aP = s_bl  + lm * D_DIM + kof;  // A[M=lm][k0+kof .. +1]
        const float* bP = s_tmp + lm * D_DIM + kof;  // B[k0+kof .. +1][N=lm] (W^T layout)

        v8f c = {0.f, 0.f, 0.f, 0.f, 0.f, 0.f, 0.f, 0.f};
#pragma unroll 8
        for (int k0 = 0; k0 < D_DIM; k0 += 4) {
            v2f a = *(const v2f*)(aP + k0);
            v2f b = *(const v2f*)(bP + k0);
            // D = A(16x4) * B(4x16) + C ; args: (neg_a, A, neg_b, B, c_mod, C, reuse_a, reuse_b)
            c = __builtin_amdgcn_wmma_f32_16x16x4_f32(false, a, false, b,
                                                      (short)0, c, false, false);
        }

        // C/D layout: VGPR r -> M = r + (lane<16 ? 0 : 8), N = lane & 15
        const int n = lm;
        if (n < NLAB) {
            const float bv = bias[n];
            const int mbase = (lane >> 4) << 3;
#pragma unroll
            for (int r = 0; r < 8; ++r) {
                int m = mbase + r;
                logits[((size_t)(bb * MW + w0 + m)) * NLAB + n] = c[r] + bv;
            }
        }
    }

    // ---- alpha = 0.5 constant ----------------------------------------
    if (tid >= 32 && tid < 32 + WBLK)
        alpha[(size_t)bb * MW + w0 + (tid - 32)] = 0.5f;
}

extern "C" void kernel_launch(void* const* d_in, const int* in_sizes, int n_in,
                              void* d_out, int out_size, void* d_ws, size_t ws_size,
                              hipStream_t stream) {
    const float* hh  = (const float*)d_in[0];   // h_hing_sub (B,S,D)
    const float* hr  = (const float*)d_in[1];   // h_rob_sub  (B,S,D)
    const int*   wih = (const int*)  d_in[2];   // hing_word_ids (B,S)
    const int*   wir = (const int*)  d_in[3];   // rob_word_ids  (B,S)
    // d_in[4] = max_words scalar (256, compile-time constant here)
    const float* W   = (const float*)d_in[5];   // (D, NLAB)
    const float* b   = (const float*)d_in[6];   // (NLAB)

    const int B = in_sizes[0] / (S_LEN * D_DIM);

    float* logits = (float*)d_out;                          // (B, MW, NLAB)
    float* alpha  = (float*)d_out + (size_t)B * MW * NLAB;  // (B, MW)

    dim3 grid(MW / WBLK, B);   // 16 word-tiles x 64 batches = 1024 blocks
    fused_align_blend_gemm<<<grid, BLOCK, 0, stream>>>(hh, hr, wih, wir, W, b,
                                                       logits, alpha);
}